// EditModel_88192858456789
// MI455X (gfx1250) — compile-verified
//
#include <hip/hip_runtime.h>
#include <math.h>

typedef __attribute__((ext_vector_type(2))) float v2f;
typedef __attribute__((ext_vector_type(4))) float v4f;
typedef __attribute__((ext_vector_type(8))) float v8f;

#define PN 32
#define MN 32
#define BN 128
#define HN 256
#define DN 54
#define ON 53
#define OPAD 64
#define CH 512  // 2H

// GEMM tiling
#define RB 128        // rows per block (8 waves x 16)
#define CB 64         // cols per block
#define KC 64         // K chunk staged in LDS
#define LDK (KC + 4)  // padded LDS row stride (68 floats) -> conflict-free ds_load_b64

// ---------------- small helper kernels ----------------

__global__ void zero_f32(float* p, int n) {
  int i = blockIdx.x * blockDim.x + threadIdx.x;
  if (i < n) p[i] = 0.f;
}

__global__ void onehot_to_idx(const float* __restrict__ oh, int* __restrict__ idx, int rows) {
  int r = blockIdx.x * blockDim.x + threadIdx.x;
  if (r >= rows) return;
  const float* p = oh + (size_t)r * DN;
  int v = 0;
  for (int d = 0; d < DN; ++d)
    if (p[d] > 0.5f) v = d;
  idx[r] = v;
}

// Pad W_sub/W_ins from [53,512] to [64,512] (zero rows) and biases to [64].
__global__ void pad_proj_weights(const float* __restrict__ Ws, const float* __restrict__ bs,
                                 const float* __restrict__ Wi, const float* __restrict__ bi,
                                 float* __restrict__ Wsp, float* __restrict__ Wip,
                                 float* __restrict__ bsp, float* __restrict__ bip) {
  int t = blockIdx.x * blockDim.x + threadIdx.x;
  if (t >= OPAD * CH) return;
  int n = t / CH;
  int k = t - n * CH;
  Wsp[t] = (n < ON) ? Ws[n * CH + k] : 0.f;
  Wip[t] = (n < ON) ? Wi[n * CH + k] : 0.f;
  if (t < OPAD) {
    bsp[t] = (t < ON) ? bs[t] : 0.f;
    bip[t] = (t < ON) ? bi[t] : 0.f;
  }
}

// ---------------- WMMA fp32 GEMM, LDS-staged weights ----------------
// out[R,N] = X[R,K] @ W[N,K]^T (+ bias[n]) (+ emb[n*embD + idx[row]])
// Block: 256 threads = 8 waves; covers 128 rows x 64 cols. Grid: (R/128, N/64).
// B (weights) double-buffered through LDS; register prefetch of next K-chunk.
// WMMA f32 16x16x4 frag layouts:
//   A (16x4): lanes 0-15 -> M=lane, K=v;  lanes 16-31 -> M=lane-16, K=2+v
//   B (4x16): lanes 0-15 -> N=lane, K=v;  lanes 16-31 -> N=lane-16, K=2+v
//   C/D:      VGPR r: lanes 0-15 -> M=r, lanes 16-31 -> M=r+8; N=lane&15
__global__ __launch_bounds__(256) void gemm_wmma_f32(
    const float* __restrict__ X, const float* __restrict__ W,
    const float* __restrict__ bias,
    const float* __restrict__ emb, const int* __restrict__ idx, int embD,
    float* __restrict__ out, int N, int K) {
  __shared__ float Blds[2][CB * LDK];

  int tid = threadIdx.x;
  int lane = tid & 31;
  int wv = tid >> 5;   // wave id 0..7 -> row strip
  int kh = lane >> 4;  // which K-half this lane holds
  int l16 = lane & 15;
  int row0 = blockIdx.x * RB + wv * 16;
  int col0 = blockIdx.y * CB;

  // B staging map: thread t covers col = t/4, k-segment = (t%4)*16 (16 floats)
  int bcol = tid >> 2;
  int bseg = (tid & 3) * 16;
  const float* Bg = W + (size_t)(col0 + bcol) * K + bseg;
  float* Bs0 = &Blds[0][0] + bcol * LDK + bseg;
  float* Bs1 = &Blds[1][0] + bcol * LDK + bseg;

  const float* Ap = X + (size_t)(row0 + l16) * K + 2 * kh;

  v8f acc0 = {}, acc1 = {}, acc2 = {}, acc3 = {};

  // ---- stage chunk 0 ----
  {
    v4f p0 = *(const v4f*)(Bg + 0);
    v4f p1 = *(const v4f*)(Bg + 4);
    v4f p2 = *(const v4f*)(Bg + 8);
    v4f p3 = *(const v4f*)(Bg + 12);
    *(v4f*)(Bs0 + 0) = p0;
    *(v4f*)(Bs0 + 4) = p1;
    *(v4f*)(Bs0 + 8) = p2;
    *(v4f*)(Bs0 + 12) = p3;
  }
  __syncthreads();

  int nchunks = K / KC;
  for (int kc = 0; kc < nchunks; ++kc) {
    // prefetch next chunk into registers (hidden behind this chunk's WMMAs)
    v4f p0, p1, p2, p3;
    bool hasNext = (kc + 1 < nchunks);
    if (hasNext) {
      const float* g = Bg + (size_t)(kc + 1) * KC;
      p0 = *(const v4f*)(g + 0);
      p1 = *(const v4f*)(g + 4);
      p2 = *(const v4f*)(g + 8);
      p3 = *(const v4f*)(g + 12);
    }

    const float* Bl = &Blds[kc & 1][0];
    const float* Al = Ap + (size_t)kc * KC;
    const float* bp0 = Bl + (0 + l16) * LDK + 2 * kh;
    const float* bp1 = Bl + (16 + l16) * LDK + 2 * kh;
    const float* bp2 = Bl + (32 + l16) * LDK + 2 * kh;
    const float* bp3 = Bl + (48 + l16) * LDK + 2 * kh;

#pragma unroll
    for (int k = 0; k < KC; k += 4) {
      v2f a = *(const v2f*)(Al + k);
      v2f b0 = *(const v2f*)(bp0 + k);
      v2f b1 = *(const v2f*)(bp1 + k);
      v2f b2 = *(const v2f*)(bp2 + k);
      v2f b3 = *(const v2f*)(bp3 + k);
      acc0 = __builtin_amdgcn_wmma_f32_16x16x4_f32(false, a, false, b0, (short)0, acc0, false, false);
      acc1 = __builtin_amdgcn_wmma_f32_16x16x4_f32(false, a, false, b1, (short)0, acc1, false, false);
      acc2 = __builtin_amdgcn_wmma_f32_16x16x4_f32(false, a, false, b2, (short)0, acc2, false, false);
      acc3 = __builtin_amdgcn_wmma_f32_16x16x4_f32(false, a, false, b3, (short)0, acc3, false, false);
    }

    if (hasNext) {
      // Write next chunk to the other buffer. Safe: its last readers finished
      // before the barrier at the end of the previous iteration.
      float* s = ((kc + 1) & 1) ? Bs1 : Bs0;
      *(v4f*)(s + 0) = p0;
      *(v4f*)(s + 4) = p1;
      *(v4f*)(s + 8) = p2;
      *(v4f*)(s + 12) = p3;
      __syncthreads();
    }
  }

  int ridx[8];
#pragma unroll
  for (int r = 0; r < 8; ++r) ridx[r] = 0;
  if (idx) {
#pragma unroll
    for (int r = 0; r < 8; ++r) ridx[r] = idx[row0 + r + kh * 8];
  }

  v8f accs[4] = {acc0, acc1, acc2, acc3};
#pragma unroll
  for (int j = 0; j < 4; ++j) {
    int col = col0 + j * 16 + l16;
    float badd = bias ? bias[col] : 0.f;
#pragma unroll
    for (int r = 0; r < 8; ++r) {
      int row = row0 + r + kh * 8;
      float v = accs[j][r] + badd;
      if (emb) v += emb[(size_t)col * embD + ridx[r]];
      out[(size_t)row * N + col] = v;
    }
  }
}

// ---------------- LSTM gate nonlinearity ----------------
// G: [B, 4*Hs] preactivations (gate order i,f,g,o). Updates c,h; also writes h
// into the time-sequence buffer at [b*CH + seqOff + u].
__global__ void lstm_gates(const float* __restrict__ G, float* __restrict__ c,
                           float* __restrict__ h, float* __restrict__ seq,
                           int Hs, int seqOff) {
  int t = blockIdx.x * blockDim.x + threadIdx.x;
  if (t >= BN * Hs) return;
  int b = t / Hs;
  int u = t - b * Hs;
  const float* g = G + (size_t)b * 4 * Hs;
  float gi = g[u];
  float gf = g[Hs + u];
  float gg = g[2 * Hs + u];
  float go = g[3 * Hs + u];
  float si = 1.f / (1.f + expf(-gi));
  float sf = 1.f / (1.f + expf(-gf));
  float so = 1.f / (1.f + expf(-go));
  float cn = sf * c[t] + si * tanhf(gg);
  float hn = so * tanhf(cn);
  c[t] = cn;
  h[t] = hn;
  seq[(size_t)b * CH + seqOff + u] = hn;
}

// ---------------- fused log-softmax + output gather ----------------
// logits_sub[p,m,b,o] = projP_sub[p,b,o] + projM_sub[m,b,o]  (distributed proj)
// p_del = lsm_sub[...,O-1]; p_end = lsm_ins[...,O-1];
// p_sub[p,m,b] (m<M-1) = lsm_sub[p,m,b,tgt[m+1,b]]  (exact: logOH trick)
__global__ void finalize(const float* __restrict__ pPs, const float* __restrict__ pPi,
                         const float* __restrict__ pMs, const float* __restrict__ pMi,
                         const int* __restrict__ tgt, float* __restrict__ out) {
  int t = blockIdx.x * blockDim.x + threadIdx.x;
  if (t >= PN * MN * BN) return;
  int b = t % BN;
  int pm = t / BN;
  int m = pm % MN;
  int p = pm / MN;
  const float* as = pPs + ((size_t)p * BN + b) * OPAD;
  const float* bsv = pMs + ((size_t)m * BN + b) * OPAD;
  const float* ai = pPi + ((size_t)p * BN + b) * OPAD;
  const float* biv = pMi + ((size_t)m * BN + b) * OPAD;
  int tg = (m < MN - 1) ? tgt[(m + 1) * BN + b] : 0;

  float mxs = -3.4e38f, mxi = -3.4e38f;
  for (int o = 0; o < ON; ++o) {
    float ls = as[o] + bsv[o];
    float li = ai[o] + biv[o];
    mxs = ls > mxs ? ls : mxs;
    mxi = li > mxi ? li : mxi;
  }
  float sums = 0.f, sumi = 0.f, ls52 = 0.f, li52 = 0.f, lst = 0.f, lit = 0.f;
  for (int o = 0; o < ON; ++o) {
    float ls = as[o] + bsv[o];
    float li = ai[o] + biv[o];
    sums += expf(ls - mxs);
    sumi += expf(li - mxi);
    if (o == ON - 1) { ls52 = ls; li52 = li; }
    if (o == tg) { lst = ls; lit = li; }
  }
  float lses = mxs + logf(sums);
  float lsei = mxi + logf(sumi);
  size_t idxPMB = ((size_t)p * MN + m) * BN + b;
  out[idxPMB] = ls52 - lses;                         // p_del
  out[(size_t)PN * MN * BN + idxPMB] = li52 - lsei;  // p_end
  if (m < MN - 1) {
    size_t base = 2ull * PN * MN * BN;
    size_t idxS = ((size_t)p * (MN - 1) + m) * BN + b;
    out[base + idxS] = lst - lses;                              // p_sub
    out[base + (size_t)PN * (MN - 1) * BN + idxS] = lit - lsei; // p_ins
  }
}

// ---------------- host launcher ----------------

extern "C" void kernel_launch(void* const* d_in, const int* in_sizes, int n_in,
                              void* d_out, int out_size, void* d_ws, size_t ws_size,
                              hipStream_t stream) {
  (void)in_sizes; (void)n_in; (void)out_size; (void)ws_size;
  const float* src_oh = (const float*)d_in[0];
  const float* tgt_oh = (const float*)d_in[1];  (void)tgt_oh; // replaced by targets_idx
  const float* Wih_pf = (const float*)d_in[2];
  const float* Whh_pf = (const float*)d_in[3];
  const float* b_pf   = (const float*)d_in[4];
  const float* Wih_pb = (const float*)d_in[5];
  const float* Whh_pb = (const float*)d_in[6];
  const float* b_pb   = (const float*)d_in[7];
  const float* Wih_m  = (const float*)d_in[8];
  const float* Whh_m  = (const float*)d_in[9];
  const float* b_m    = (const float*)d_in[10];
  const float* W_sub  = (const float*)d_in[11];
  const float* b_sub  = (const float*)d_in[12];
  const float* W_ins  = (const float*)d_in[13];
  const float* b_ins  = (const float*)d_in[14];
  const int*   tgt    = (const int*)d_in[15];

  char* wptr = (char*)d_ws;
  auto alloc = [&](size_t nfloats) -> float* {
    float* p = (float*)wptr;
    wptr += ((nfloats * 4 + 255) / 256) * 256;
    return p;
  };
  float* prior  = alloc((size_t)PN * BN * CH);  // [P,B,512] fwd(0:256)|bwd(256:512)
  float* modern = alloc((size_t)MN * BN * CH);  // [M,B,512]
  float* Gpf = alloc((size_t)BN * 4 * HN);
  float* Gpb = alloc((size_t)BN * 4 * HN);
  float* Gm  = alloc((size_t)BN * 4 * CH);
  int hcN = 4 * BN * HN + 2 * BN * CH;
  float* hc = alloc(hcN);
  float* h_pf = hc;
  float* c_pf = h_pf + BN * HN;
  float* h_pb = c_pf + BN * HN;
  float* c_pb = h_pb + BN * HN;
  float* h_m  = c_pb + BN * HN;
  float* c_m  = h_m + BN * CH;
  float* Wsp = alloc(OPAD * CH);
  float* Wip = alloc(OPAD * CH);
  float* bsp = alloc(OPAD);
  float* bip = alloc(OPAD);
  float* pPs = alloc((size_t)PN * BN * OPAD);
  float* pPi = alloc((size_t)PN * BN * OPAD);
  float* pMs = alloc((size_t)MN * BN * OPAD);
  float* pMi = alloc((size_t)MN * BN * OPAD);
  int* sidx = (int*)alloc(PN * BN);

  zero_f32<<<(hcN + 255) / 256, 256, 0, stream>>>(hc, hcN);
  onehot_to_idx<<<(PN * BN + 255) / 256, 256, 0, stream>>>(src_oh, sidx, PN * BN);
  pad_proj_weights<<<(OPAD * CH + 255) / 256, 256, 0, stream>>>(
      W_sub, b_sub, W_ins, b_ins, Wsp, Wip, bsp, bip);

  for (int t = 0; t < PN; ++t) {
    // prior forward: G = h@Whh^T + Wih[:,src_idx] + b
    gemm_wmma_f32<<<dim3(1, (4 * HN) / CB), 256, 0, stream>>>(
        h_pf, Whh_pf, b_pf, Wih_pf, sidx + t * BN, DN, Gpf, 4 * HN, HN);
    lstm_gates<<<(BN * HN + 255) / 256, 256, 0, stream>>>(
        Gpf, c_pf, h_pf, prior + (size_t)t * BN * CH, HN, 0);
    // prior backward (consumes reversed sequence, stores reversed back)
    int tb = PN - 1 - t;
    gemm_wmma_f32<<<dim3(1, (4 * HN) / CB), 256, 0, stream>>>(
        h_pb, Whh_pb, b_pb, Wih_pb, sidx + tb * BN, DN, Gpb, 4 * HN, HN);
    lstm_gates<<<(BN * HN + 255) / 256, 256, 0, stream>>>(
        Gpb, c_pb, h_pb, prior + (size_t)tb * BN * CH, HN, HN);
    // modern (hidden 2H)
    gemm_wmma_f32<<<dim3(1, (4 * CH) / CB), 256, 0, stream>>>(
        h_m, Whh_m, b_m, Wih_m, tgt + t * BN, DN, Gm, 4 * CH, CH);
    lstm_gates<<<(BN * CH + 255) / 256, 256, 0, stream>>>(
        Gm, c_m, h_m, modern + (size_t)t * BN * CH, CH, 0);
  }

  // Distributed projections: [P*B,512]@[64,512]^T and [M*B,512]@[64,512]^T
  gemm_wmma_f32<<<dim3(PN * BN / RB, 1), 256, 0, stream>>>(
      prior, Wsp, bsp, nullptr, nullptr, 0, pPs, OPAD, CH);
  gemm_wmma_f32<<<dim3(PN * BN / RB, 1), 256, 0, stream>>>(
      prior, Wip, bip, nullptr, nullptr, 0, pPi, OPAD, CH);
  gemm_wmma_f32<<<dim3(MN * BN / RB, 1), 256, 0, stream>>>(
      modern, Wsp, bsp, nullptr, nullptr, 0, pMs, OPAD, CH);
  gemm_wmma_f32<<<dim3(MN * BN / RB, 1), 256, 0, stream>>>(
      modern, Wip, bip, nullptr, nullptr, 0, pMi, OPAD, CH);

  finalize<<<(PN * MN * BN + 255) / 256, 256, 0, stream>>>(
      pPs, pPi, pMs, pMi, tgt, (float*)d_out);
}